// ToySingle_816043786390
// MI455X (gfx1250) — compile-verified
//
#include <hip/hip_runtime.h>

typedef float v2f __attribute__((ext_vector_type(2)));
typedef float v8f __attribute__((ext_vector_type(8)));

#define F 100
#define FP 112          // F padded to 7*16
#define KP2 (FP / 2)    // 56 K-pairs (K padded to 112; pairs >= 50 are zero)
#define NTILES 7
#define BLOCK_THREADS 512
#define WAVES_PER_BLOCK (BLOCK_THREADS / 32)
#define ROWS_PER_BLOCK (WAVES_PER_BLOCK * 16)   // 256

// Kernel 1: a = x @ W + b  (rows = 4*N total), written to BOTH out and acopy.
// One wave computes a 16x112 tile via V_WMMA_F32_16X16X4_F32, K looped 25x.
// W is staged in LDS in fragment-native pair layout: WP2[k2][col][2] =
// (W[2*k2][col], W[2*k2+1][col]) so each B fragment is one aligned ds_load_b64.
__global__ __launch_bounds__(BLOCK_THREADS)
void gemm_bias_kernel(const float* __restrict__ x,
                      const float* __restrict__ W,
                      const float* __restrict__ bias,
                      float* __restrict__ out,
                      float* __restrict__ acopy,
                      long long totalRows) {
    __shared__ float WP2[KP2 * FP * 2];   // 50176 B

    const int t = threadIdx.x;
    // Stage W -> LDS, zero padded, pairwise along K.
    for (int i = t; i < KP2 * FP; i += BLOCK_THREADS) {
        const int k2 = i / FP, col = i % FP;
        const int k = 2 * k2;
        float lo = 0.0f, hi = 0.0f;
        if (col < F) {
            if (k < F)     lo = W[k * F + col];
            if (k + 1 < F) hi = W[(k + 1) * F + col];
        }
        WP2[i * 2 + 0] = lo;
        WP2[i * 2 + 1] = hi;
    }
    __syncthreads();

    const int wave = t >> 5;            // wave32
    const int lane = t & 31;
    const int lh   = lane >> 4;         // lane half (0/1)
    const int l16  = lane & 15;

    const long long rowBase = (long long)blockIdx.x * ROWS_PER_BLOCK + (long long)wave * 16;
    long long arow = rowBase + l16;
    if (arow >= totalRows) arow = totalRows - 1;
    const float* __restrict__ xrow = x + arow * F;

    // Per-lane base into the pair layout: k2 = k0/2 + lh, element (k2, col).
    const float* __restrict__ wbase = &WP2[((lh)*FP + l16) * 2];

    v8f acc[NTILES];
#pragma unroll
    for (int nt = 0; nt < NTILES; ++nt) acc[nt] = (v8f){0,0,0,0,0,0,0,0};

    // K loop: 25 steps of 4
    for (int k0 = 0; k0 < F; k0 += 4) {
        // A 16x4 f32 fragment: lane l holds (M=l%16, K=k0+2*lh + {0,1}) -> one b64 load
        const v2f afrag = *(const v2f*)(xrow + k0 + 2 * lh);
        const float* __restrict__ wk = wbase + (k0 >> 1) * (FP * 2);
#pragma unroll
        for (int nt = 0; nt < NTILES; ++nt) {
            // B 4x16 f32 fragment: vgpr j, lane l -> (K=k0+2*lh+j, N=nt*16+l%16):
            // both elements adjacent in LDS -> single ds_load_b64, imm offset nt*128B
            const v2f bfrag = *(const v2f*)(wk + nt * 32);
            acc[nt] = __builtin_amdgcn_wmma_f32_16x16x4_f32(
                false, afrag, false, bfrag, (short)0, acc[nt], false, false);
        }
    }

    // C/D layout: vgpr i, lane -> M = i + 8*lh, N = l16. Bias is scalar per (lane, ntile).
#pragma unroll
    for (int nt = 0; nt < NTILES; ++nt) {
        const int ncol = nt * 16 + l16;
        if (ncol < F) {
            const float bv = bias[ncol];
#pragma unroll
            for (int i = 0; i < 8; ++i) {
                const long long row = rowBase + i + 8 * lh;
                if (row < totalRows) {
                    const float v = acc[nt][i] + bv;
                    out[row * F + ncol]   = v;
                    acopy[row * F + ncol] = v;
                }
            }
        }
    }
}

// Kernel 2: for 12 (dst,src) pairs: out[dst, from_idx[dst,src,r], f] += a[src, to_idx[src,dst,r], f]
// Gathers read the pristine copy; scatter uses non-returning global_atomic_add_f32.
__global__ __launch_bounds__(256)
void shuffle_scatter_kernel(const float* __restrict__ a,
                            const long long* __restrict__ to_idx,
                            const long long* __restrict__ from_idx,
                            float* __restrict__ out,
                            long long nPerDev, long long M, long long total) {
    const long long stride = (long long)gridDim.x * blockDim.x;
    for (long long gid = (long long)blockIdx.x * blockDim.x + threadIdx.x;
         gid < total; gid += stride) {
        const int f = (int)(gid % F);
        const long long tr = gid / F;
        const long long r = tr % M;
        const int p = (int)(tr / M);           // 0..11
        const int dst = p / 3;
        const int s3  = p - dst * 3;           // 0..2
        const int src = s3 + (s3 >= dst ? 1 : 0);

        const long long ti = to_idx[((long long)src * 4 + dst) * M + r];
        const long long fi = from_idx[((long long)dst * 4 + src) * M + r];
        const float v = a[((long long)src * nPerDev + ti) * F + f];
        atomicAdd(&out[((long long)dst * nPerDev + fi) * F + f], v);
    }
}

extern "C" void kernel_launch(void* const* d_in, const int* in_sizes, int n_in,
                              void* d_out, int out_size, void* d_ws, size_t ws_size,
                              hipStream_t stream) {
    const float*     x        = (const float*)d_in[0];
    const float*     W        = (const float*)d_in[1];
    const float*     bias     = (const float*)d_in[2];
    const long long* to_idx   = (const long long*)d_in[3];
    const long long* from_idx = (const long long*)d_in[4];
    float* out = (float*)d_out;

    const long long totalRows = (long long)in_sizes[0] / F;   // 4*N = 524288
    const long long nPerDev   = totalRows / 4;                // N = 131072
    const long long M         = (long long)in_sizes[3] / 16;  // 32768

    // Pristine copy of `a` for the gather side (ws if large enough, else best-effort out)
    const size_t needed = (size_t)totalRows * F * sizeof(float);
    float* acopy = (ws_size >= needed) ? (float*)d_ws : out;

    const int gemmBlocks = (int)((totalRows + ROWS_PER_BLOCK - 1) / ROWS_PER_BLOCK);
    gemm_bias_kernel<<<gemmBlocks, BLOCK_THREADS, 0, stream>>>(
        x, W, bias, out, acopy, totalRows);

    const long long total = 12LL * M * F;
    int blocks = (int)((total + 255) / 256);
    if (blocks > 262144) blocks = 262144;
    shuffle_scatter_kernel<<<blocks, 256, 0, stream>>>(
        (const float*)acopy, to_idx, from_idx, out, nPerDev, M, total);
}